// FedG_DQN_3307124818434
// MI455X (gfx1250) — compile-verified
//
#include <hip/hip_runtime.h>
#include <hip/hip_bf16.h>
#include <stdint.h>

// ---------------------------------------------------------------------------
// GINEConv x2 + Q-head for MI455X (gfx1250).
//   Edge phase: bandwidth-bound gather + relu + scatter (global_atomic_add_f32).
//   Node MLPs:  fp32 WMMA (V_WMMA_F32_16X16X4_F32), 16-node tiles, 4 waves/block.
// ---------------------------------------------------------------------------

typedef float v2f __attribute__((ext_vector_type(2)));
typedef float v8f __attribute__((ext_vector_type(8)));

#define HID 64

__device__ __forceinline__ void atomic_add_f32_g(float* p, float v) {
    unsigned long long a = (unsigned long long)(uintptr_t)p;
    asm volatile("global_atomic_add_f32 %0, %1, off" :: "v"(a), "v"(v) : "memory");
}

__global__ __launch_bounds__(256) void zero_f32(float* __restrict__ p, int n) {
    int i = blockIdx.x * 256 + threadIdx.x;
    if (i < n) p[i] = 0.0f;
}

// msg = relu(x[src] + edge_attr[e]*We + be);  agg[dst] += msg   (per feature)
// one thread handles 4 features of one edge (float4 gather, 4 f32 atomics).
__global__ __launch_bounds__(256) void edge_msg(
    const float* __restrict__ xin, const float* __restrict__ eattr,
    const int* __restrict__ src, const int* __restrict__ dst,
    const float* __restrict__ We, const float* __restrict__ be,
    float* __restrict__ agg, int nedge)
{
    long long idx = (long long)blockIdx.x * 256 + threadIdx.x;
    long long total = (long long)nedge * (HID / 4);
    if (idx >= total) return;
    int e = (int)(idx >> 4);          // HID/4 == 16 threads per edge
    int f = ((int)idx & 15) * 4;
    int s = src[e];
    int d = dst[e];
    float ea = eattr[e];
    const float4 xv = *reinterpret_cast<const float4*>(xin + (long long)s * HID + f);
    const float4 wv = *reinterpret_cast<const float4*>(We + f);
    const float4 bv = *reinterpret_cast<const float4*>(be + f);
    float m0 = fmaxf(fmaf(ea, wv.x, xv.x + bv.x), 0.0f);
    float m1 = fmaxf(fmaf(ea, wv.y, xv.y + bv.y), 0.0f);
    float m2 = fmaxf(fmaf(ea, wv.z, xv.z + bv.z), 0.0f);
    float m3 = fmaxf(fmaf(ea, wv.w, xv.w + bv.w), 0.0f);
    float* base = agg + (long long)d * HID + f;
    atomic_add_f32_g(base + 0, m0);
    atomic_add_f32_g(base + 1, m1);
    atomic_add_f32_g(base + 2, m2);
    atomic_add_f32_g(base + 3, m3);
}

// Fused GINE node MLP:  out = relu?( relu((x+agg) @ Wa + ba) @ Wb + bb )
// 16-node tile per block; 4 waves; wave w computes output cols [16w,16w+16).
// fp32 WMMA 16x16x4: A lane L -> A[L%16][k0+(L/16)*2+j]; B lane L -> B[k][L%16];
// C lane L, reg v -> C[v+(L/16)*8][L%16].
__global__ __launch_bounds__(128) void gine_mlp(
    const float* __restrict__ xin, const float* __restrict__ agg,
    const float* __restrict__ Wa, const float* __restrict__ ba,
    const float* __restrict__ Wb, const float* __restrict__ bb,
    float* __restrict__ out, int nnode, int relu_out)
{
    __shared__ float sWa[HID * HID];
    __shared__ float sWb[HID * HID];
    __shared__ float sIn[16 * 68];   // stride 68 -> conflict-free column reads
    __shared__ float sT[16 * 68];

    const int tid   = threadIdx.x;
    const int node0 = blockIdx.x * 16;

    for (int i = tid; i < HID * HID; i += 128) { sWa[i] = Wa[i]; sWb[i] = Wb[i]; }
    for (int i = tid; i < 16 * HID; i += 128) {
        int r = i >> 6, f = i & 63;
        int n = min(node0 + r, nnode - 1);
        sIn[r * 68 + f] = xin[(long long)n * HID + f] + agg[(long long)n * HID + f];
    }
    __syncthreads();

    const int wave  = tid >> 5;
    const int lane  = tid & 31;
    const int mrow  = lane & 15;
    const int khalf = (lane >> 4) * 2;          // 0 or 2
    const int mbase = (lane >> 4) * 8;          // C row base for this half
    const int ncol  = wave * 16 + mrow;

    v8f c = {};
    #pragma unroll
    for (int k0 = 0; k0 < HID; k0 += 4) {
        v2f a, b;
        a.x = sIn[mrow * 68 + k0 + khalf];
        a.y = sIn[mrow * 68 + k0 + khalf + 1];
        b.x = sWa[(k0 + khalf) * HID + ncol];
        b.y = sWa[(k0 + khalf + 1) * HID + ncol];
        c = __builtin_amdgcn_wmma_f32_16x16x4_f32(false, a, false, b,
                                                  (short)0, c, false, false);
    }
    float biasA = ba[ncol];
    #pragma unroll
    for (int v = 0; v < 8; v++) {
        float t = c[v] + biasA;
        t = t > 0.0f ? t : 0.0f;
        sT[(v + mbase) * 68 + ncol] = t;
    }
    __syncthreads();

    v8f c2 = {};
    #pragma unroll
    for (int k0 = 0; k0 < HID; k0 += 4) {
        v2f a, b;
        a.x = sT[mrow * 68 + k0 + khalf];
        a.y = sT[mrow * 68 + k0 + khalf + 1];
        b.x = sWb[(k0 + khalf) * HID + ncol];
        b.y = sWb[(k0 + khalf + 1) * HID + ncol];
        c2 = __builtin_amdgcn_wmma_f32_16x16x4_f32(false, a, false, b,
                                                   (short)0, c2, false, false);
    }
    float biasB = bb[ncol];
    #pragma unroll
    for (int v = 0; v < 8; v++) {
        float t = c2[v] + biasB;
        if (relu_out) t = t > 0.0f ? t : 0.0f;
        int m = node0 + v + mbase;
        if (m < nnode) out[(long long)m * HID + ncol] = t;
    }
}

// Q head: cat[K=16,192] = [curr | dest | nbr]; q = relu(cat@Wl1+bl1)@Wl2+bl2
__global__ __launch_bounds__(128) void qhead(
    const float* __restrict__ h,
    const int* __restrict__ curr, const int* __restrict__ dest,
    const int* __restrict__ nbr,
    const float* __restrict__ Wl1, const float* __restrict__ bl1,
    const float* __restrict__ Wl2, const float* __restrict__ bl2,
    float* __restrict__ out)
{
    __shared__ float sCat[16 * 196];  // stride 196: (4r+c)%64 distinct per column
    __shared__ float sT[16 * 68];

    const int tid = threadIdx.x;
    const int ci = curr[0], di = dest[0];
    for (int i = tid; i < 16 * 192; i += 128) {
        int r = i / 192, cc = i - r * 192;
        int node, f;
        if (cc < 64)        { node = ci;     f = cc; }
        else if (cc < 128)  { node = di;     f = cc - 64; }
        else                { node = nbr[r]; f = cc - 128; }
        sCat[r * 196 + cc] = h[(long long)node * HID + f];
    }
    __syncthreads();

    const int wave  = tid >> 5;
    const int lane  = tid & 31;
    const int mrow  = lane & 15;
    const int khalf = (lane >> 4) * 2;
    const int mbase = (lane >> 4) * 8;
    const int ncol  = wave * 16 + mrow;

    v8f c = {};
    #pragma unroll 4
    for (int k0 = 0; k0 < 192; k0 += 4) {
        v2f a, b;
        a.x = sCat[mrow * 196 + k0 + khalf];
        a.y = sCat[mrow * 196 + k0 + khalf + 1];
        b.x = Wl1[(k0 + khalf) * HID + ncol];
        b.y = Wl1[(k0 + khalf + 1) * HID + ncol];
        c = __builtin_amdgcn_wmma_f32_16x16x4_f32(false, a, false, b,
                                                  (short)0, c, false, false);
    }
    float bias = bl1[ncol];
    #pragma unroll
    for (int v = 0; v < 8; v++) {
        float t = c[v] + bias;
        t = t > 0.0f ? t : 0.0f;
        sT[(v + mbase) * 68 + ncol] = t;
    }
    __syncthreads();

    if (tid < 16) {
        float acc = bl2[0];
        #pragma unroll 8
        for (int f = 0; f < HID; f++) acc += sT[tid * 68 + f] * Wl2[f];
        out[tid] = acc;
    }
}

extern "C" void kernel_launch(void* const* d_in, const int* in_sizes, int n_in,
                              void* d_out, int out_size, void* d_ws, size_t ws_size,
                              hipStream_t stream) {
    const float* x     = (const float*)d_in[0];
    const int*   eidx  = (const int*)d_in[1];
    const int*   curr  = (const int*)d_in[2];
    const int*   dest  = (const int*)d_in[3];
    const int*   nbr   = (const int*)d_in[4];
    const float* eattr = (const float*)d_in[5];
    const float* We1 = (const float*)d_in[6],  *be1 = (const float*)d_in[7];
    const float* W1a = (const float*)d_in[8],  *b1a = (const float*)d_in[9];
    const float* W1b = (const float*)d_in[10], *b1b = (const float*)d_in[11];
    const float* We2 = (const float*)d_in[12], *be2 = (const float*)d_in[13];
    const float* W2a = (const float*)d_in[14], *b2a = (const float*)d_in[15];
    const float* W2b = (const float*)d_in[16], *b2b = (const float*)d_in[17];
    const float* Wl1 = (const float*)d_in[18], *bl1 = (const float*)d_in[19];
    const float* Wl2 = (const float*)d_in[20], *bl2 = (const float*)d_in[21];

    const int nedge = in_sizes[5];            // E (edge_attr has E elements)
    const int nnode = in_sizes[0] / HID;      // N
    const int* srcI = eidx;
    const int* dstI = eidx + nedge;

    float* agg = (float*)d_ws;
    float* h1  = agg + (size_t)nnode * HID;
    float* h2  = h1  + (size_t)nnode * HID;

    const int nfeat = nnode * HID;
    const long long ethreads = (long long)nedge * (HID / 4);
    const int eblocks = (int)((ethreads + 255) / 256);
    const int mblocks = (nnode + 15) / 16;

    // ---- layer 1 ----
    zero_f32<<<(nfeat + 255) / 256, 256, 0, stream>>>(agg, nfeat);
    edge_msg<<<eblocks, 256, 0, stream>>>(x, eattr, srcI, dstI, We1, be1, agg, nedge);
    gine_mlp<<<mblocks, 128, 0, stream>>>(x, agg, W1a, b1a, W1b, b1b, h1, nnode, 1);

    // ---- layer 2 ----
    zero_f32<<<(nfeat + 255) / 256, 256, 0, stream>>>(agg, nfeat);
    edge_msg<<<eblocks, 256, 0, stream>>>(h1, eattr, srcI, dstI, We2, be2, agg, nedge);
    gine_mlp<<<mblocks, 128, 0, stream>>>(h1, agg, W2a, b2a, W2b, b2b, h2, nnode, 0);

    // ---- Q head ----
    qhead<<<1, 128, 0, stream>>>(h2, curr, dest, nbr, Wl1, bl1, Wl2, bl2, (float*)d_out);
}